// TaximTorch_77730318123538
// MI455X (gfx1250) — compile-verified
//
#include <hip/hip_runtime.h>
#include <cstdint>
#include <cstddef>

// ---------------------------------------------------------------------------
// TaximTorch fused pipeline for MI455X (gfx1250).
// Memory-bound (~85 MB HBM vs a few GFLOP) -> no WMMA; use the CDNA5 async
// global->LDS path (GLOBAL_LOAD_ASYNC_TO_LDS_B32 + s_wait_asynccnt) to stage
// blur tiles, then do everything (blur/gradient/binning/poly/clip) in one pass.
// ---------------------------------------------------------------------------

#if defined(__gfx1250__) && __has_builtin(__builtin_amdgcn_global_load_async_to_lds_b32)
#define HAVE_ASYNC_LDS 1
#else
#define HAVE_ASYNC_LDS 0
#endif

namespace {
constexpr int Bn = 16, H = 480, W = 640, NB = 120, K = 21, RAD = 10;
constexpr int TH = 48, TW = 32;          // output tile per workgroup
constexpr int IH = TH + 2 * RAD + 2;     // 70  (blur halo + gradient halo)
constexpr int IW = TW + 2 * RAD + 2;     // 54
constexpr int LPIT = IW + 1;             // 55  (bank-conflict pad)
constexpr int MW = TW + 2;               // 34  (row-blurred width)
constexpr int MPIT = MW + 1;             // 35
constexpr int BH = TH + 2;               // 50  (blurred rows incl. grad halo)
constexpr int NTHREADS = 256;
constexpr float PIXMM_C = 0.0295f;
constexpr float PI_C = 3.14159265358979323846f;
}

__device__ __forceinline__ int iclamp(int v, int lo, int hi) {
  return v < lo ? lo : (v > hi ? hi : v);
}

// Async global->LDS copy of one dword (CDNA5 GLOBAL_LOAD_ASYNC_TO_LDS_B32).
// Builtin signature (from hipcc diagnostic): (AS1 int*, AS3 int*, int, int).
__device__ __forceinline__ void async_cp_f32(float* lds_dst, const float* g_src) {
#if HAVE_ASYNC_LDS
  typedef __attribute__((address_space(3))) int lds_int_t;
  typedef __attribute__((address_space(1))) int gbl_int_t;
  // Generic pointer to a __shared__ object: low 32 bits are the LDS offset.
  lds_int_t* l = (lds_int_t*)(unsigned)(unsigned long long)lds_dst;
  gbl_int_t* g = (gbl_int_t*)(unsigned long long)g_src;
  __builtin_amdgcn_global_load_async_to_lds_b32(g, l, /*offset=*/0, /*cpol=*/0);
#else
  *lds_dst = *g_src;
#endif
}

__device__ __forceinline__ void wait_async_all() {
#if HAVE_ASYNC_LDS
#if __has_builtin(__builtin_amdgcn_s_wait_asynccnt)
  __builtin_amdgcn_s_wait_asynccnt(0);
#else
#if defined(__gfx1250__)
  asm volatile("s_wait_asynccnt 0" ::: "memory");
#endif
#endif
#endif
}

__global__ __launch_bounds__(NTHREADS) void taxim_fused_kernel(
    const float* __restrict__ hm,    // [B,H,W]
    const float* __restrict__ gel,   // [H,W]
    const float* __restrict__ poly,  // [3,NB,NB,6]
    const float* __restrict__ bg,    // [3,H,W]
    const float* __restrict__ ker,   // [K,K] separable gaussian
    float* __restrict__ out)         // [B,3,H,W]
{
  __shared__ float sH[IH * LPIT];     // height tile (later: max(height,gel))
  __shared__ float sG[IH * LPIT];     // gel tile
  __shared__ float sMid[IH * MPIT];   // after horizontal blur pass
  __shared__ float sBlur[BH * MPIT];  // after vertical blur pass
  __shared__ float sGn[K];            // 1-D gaussian weights

  const int tid = (int)threadIdx.x;
  const int ox = (int)blockIdx.x * TW;
  const int oy = (int)blockIdx.y * TH;
  const int b  = (int)blockIdx.z;
  const float* hmb = hm + (size_t)b * (size_t)(H * W);

  // --- Issue async tile staging FIRST so the DMA overlaps the weight prep ---
  // Clamped (edge-replicate) addressing == reference's mode='edge' padding.
  for (int idx = tid; idx < IH * IW; idx += NTHREADS) {
    const int r = idx / IW, c = idx - r * IW;
    const int gr = iclamp(oy - RAD - 1 + r, 0, H - 1);
    const int gc = iclamp(ox - RAD - 1 + c, 0, W - 1);
    const size_t off = (size_t)gr * W + gc;
    async_cp_f32(&sH[r * LPIT + c], hmb + off);
    async_cp_f32(&sG[r * LPIT + c], gel + off);
  }

  // 1-D weights: kernel = gn (x) gn with sum(gn)=1  =>  gn[i] = sum_j K2d[i][j]
  // (runs while the async DMA is in flight)
  if (tid < K) {
    float s = 0.f;
#pragma unroll
    for (int j = 0; j < K; ++j) s += ker[tid * K + j];
    sGn[tid] = s;
  }

  wait_async_all();       // this wave's ASYNCcnt -> 0 (its LDS writes landed)
  __syncthreads();        // all waves' async data + sGn visible

  // zq = max(height_map, gel_map), in place
  for (int idx = tid; idx < IH * IW; idx += NTHREADS) {
    const int r = idx / IW, c = idx - r * IW;
    sH[r * LPIT + c] = fmaxf(sH[r * LPIT + c], sG[r * LPIT + c]);
  }
  __syncthreads();

  // cache 1-D weights in registers (avoid 21 LDS reads per output element)
  float wr[K];
#pragma unroll
  for (int t = 0; t < K; ++t) wr[t] = sGn[t];

  // --- Horizontal blur pass ---
  for (int idx = tid; idx < IH * MW; idx += NTHREADS) {
    const int r = idx / MW, i = idx - r * MW;
    const float* src = &sH[r * LPIT + i];
    float acc = 0.f;
#pragma unroll
    for (int t = 0; t < K; ++t) acc = fmaf(src[t], wr[t], acc);
    sMid[r * MPIT + i] = acc;
  }
  __syncthreads();

  // --- Vertical blur pass ---
  for (int idx = tid; idx < BH * MW; idx += NTHREADS) {
    const int j = idx / MW, i = idx - j * MW;
    const float* src = &sMid[j * MPIT + i];
    float acc = 0.f;
#pragma unroll
    for (int t = 0; t < K; ++t) acc = fmaf(src[t * MPIT], wr[t], acc);
    sBlur[j * MPIT + i] = acc;
  }
  __syncthreads();

  // --- Gradient -> binning -> poly_grad gather -> polynomial + bg -> clip ---
  const float gscale = -1.0f / PIXMM_C;               // d(hpx) = gscale * d(blur)
  const float inv_xb = (float)(NB - 1) / (0.5f * PI_C);
  const float inv_yb = (float)(NB - 1) / (2.0f * PI_C);

  for (int idx = tid; idx < TH * TW; idx += NTHREADS) {
    const int y = idx / TW, x = idx - y * TW;
    const int gr = oy + y, gc = ox + x;
    const int j = y + 1, i = x + 1;

    const float c00 = sBlur[j * MPIT + i];
    float gy_, gx_;
    if (gr == 0)            gy_ = sBlur[(j + 1) * MPIT + i] - c00;
    else if (gr == H - 1)   gy_ = c00 - sBlur[(j - 1) * MPIT + i];
    else                    gy_ = 0.5f * (sBlur[(j + 1) * MPIT + i] -
                                          sBlur[(j - 1) * MPIT + i]);
    if (gc == 0)            gx_ = sBlur[j * MPIT + i + 1] - c00;
    else if (gc == W - 1)   gx_ = c00 - sBlur[j * MPIT + i - 1];
    else                    gx_ = 0.5f * (sBlur[j * MPIT + i + 1] -
                                          sBlur[j * MPIT + i - 1]);
    gy_ *= gscale;
    gx_ *= gscale;

    const float mag = atanf(sqrtf(fmaf(gx_, gx_, gy_ * gy_)));
    const float dir = atan2f(gy_, gx_);
    const int im = iclamp((int)floorf(mag * inv_xb), 0, NB - 1);
    const int id = iclamp((int)floorf((dir + PI_C) * inv_yb), 0, NB - 1);

    const float xf = (float)gc, yf = (float)gr;
    const float f0 = xf * xf, f1 = yf * yf, f2 = xf * yf;
    const int base = (im * NB + id) * 6;
#pragma unroll
    for (int c = 0; c < 3; ++c) {
      const float* p = poly + (size_t)c * (NB * NB * 6) + base;
      float s = fmaf(p[0], f0,
                fmaf(p[1], f1,
                fmaf(p[2], f2,
                fmaf(p[3], xf,
                fmaf(p[4], yf, p[5])))));
      s += bg[((size_t)c * H + gr) * W + gc];
      s = fminf(fmaxf(s, 0.0f), 1.0f);
      out[(((size_t)b * 3 + c) * H + gr) * W + gc] = s;
    }
  }
}

extern "C" void kernel_launch(void* const* d_in, const int* in_sizes, int n_in,
                              void* d_out, int out_size, void* d_ws, size_t ws_size,
                              hipStream_t stream) {
  (void)in_sizes; (void)n_in; (void)out_size; (void)d_ws; (void)ws_size;
  const float* hm  = (const float*)d_in[0];  // height_map [16,480,640]
  const float* gel = (const float*)d_in[1];  // gel_map    [480,640]
  const float* pg  = (const float*)d_in[2];  // poly_grad  [3,120,120,6]
  const float* bg  = (const float*)d_in[3];  // bg_proc    [3,480,640]
  const float* ker = (const float*)d_in[4];  // kernel     [21,21]
  float* out = (float*)d_out;                // [16,3,480,640]

  dim3 grid(W / TW, H / TH, Bn);             // 20 x 10 x 16 workgroups
  taxim_fused_kernel<<<grid, NTHREADS, 0, stream>>>(hm, gel, pg, bg, ker, out);
}